// BeamSearch_4612794876740
// MI455X (gfx1250) — compile-verified
//
#include <hip/hip_runtime.h>
#include <cstdint>

#define BB 32
#define LL 128
#define VV 32000
#define KK 3
#define EPSF 2.220446049250313e-16f
#define NEGF (-1.0e30f)

#define THREADS 256
#define VEC4_PER_ROW (VV / 4)                           /* 8000 */
#define NCHUNK ((VEC4_PER_ROW + THREADS - 1) / THREADS) /* 32 */
#define NBUF 4

// Branchless insert of (v,i) into a sorted top-3 with jax tie-break
// (value desc, index asc). For a sorted tie-indexed list, b0 => b1 => b2.
__device__ __forceinline__ void ins_top3(float& v0, float& v1, float& v2,
                                         unsigned& i0, unsigned& i1, unsigned& i2,
                                         float v, unsigned i) {
    const bool b0 = (v > v0) | ((v == v0) & (i < i0));
    const bool b1 = (v > v1) | ((v == v1) & (i < i1));
    const bool b2 = (v > v2) | ((v == v2) & (i < i2));
    const float    nv2 = b1 ? v1 : (b2 ? v : v2);
    const unsigned ni2 = b1 ? i1 : (b2 ? i : i2);
    const float    nv1 = b0 ? v0 : (b1 ? v : v1);
    const unsigned ni1 = b0 ? i0 : (b1 ? i : i1);
    const float    nv0 = b0 ? v : v0;
    const unsigned ni0 = b0 ? i : i0;
    v0 = nv0; v1 = nv1; v2 = nv2;
    i0 = ni0; i1 = ni1; i2 = ni2;
}

// Branchless online-softmax + top-3 step; exactly one v_exp_f32 per element.
__device__ __forceinline__ void online_step(float& m, float& s,
                                            float& v0, float& v1, float& v2,
                                            unsigned& i0, unsigned& i1, unsigned& i2,
                                            float xv, unsigned idx) {
    const bool  gt = xv > m;
    const float nm = gt ? xv : m;
    const float e  = __expf((gt ? m : xv) - nm);
    s = fmaf(s, gt ? e : 1.0f, gt ? 1.0f : e);
    m = nm;
    ins_top3(v0, v1, v2, i0, i1, i2, xv, idx);
}

// Stage 1: one block per (b,l) row. Stream 32000 floats through a 4-deep
// async global->LDS pipeline (ASYNCcnt, 3 chunks in flight per wave);
// two independent accumulator sets per lane for ILP/dual-issue.
__global__ __launch_bounds__(THREADS) void rowstats_kernel(
    const float* __restrict__ logits, float* __restrict__ ws) {
    const int row = blockIdx.x;       // 0 .. B*L-1
    const int tid = threadIdx.x;
    const unsigned long long rowBase =
        (unsigned long long)(logits + (size_t)row * VV);

    __shared__ __align__(16) float stage[NBUF][THREADS * 4];   // 16 KB
    __shared__ float red_m[THREADS], red_s[THREADS];
    __shared__ float red_v[3][THREADS];
    __shared__ unsigned red_i[3][THREADS];

    // Two accumulator sets (even/odd elements within each float4).
    float mA = -3.4e38f, sA = 0.0f, mB = -3.4e38f, sB = 0.0f;
    float a0 = -3.4e38f, a1 = -3.4e38f, a2 = -3.4e38f;
    float b0 = -3.4e38f, b1 = -3.4e38f, b2 = -3.4e38f;
    unsigned ai0 = 0xFFFFFFFFu, ai1 = 0xFFFFFFFFu, ai2 = 0xFFFFFFFFu;
    unsigned bi0 = 0xFFFFFFFFu, bi1 = 0xFFFFFFFFu, bi2 = 0xFFFFFFFFu;

    // Every wave issues every chunk (uniform ASYNCcnt across waves).
    // Out-of-range lanes clamp the global offset; their LDS region is
    // written with valid-but-unused data and never read.
    auto issue = [&](int c) {
        const int v4  = c * THREADS + tid;
        const int cv4 = v4 < VEC4_PER_ROW ? v4 : (VEC4_PER_ROW - 1);
        unsigned lds  = (unsigned)(uintptr_t)&stage[c & (NBUF - 1)][tid * 4];
        unsigned goff = (unsigned)(cv4 * 16);
        asm volatile("global_load_async_to_lds_b128 %0, %1, %2"
                     :: "v"(lds), "v"(goff), "s"(rowBase) : "memory");
    };

    auto process = [&](int c) {
        const int v4 = c * THREADS + tid;
        if (v4 < VEC4_PER_ROW) {
            const float4 x = *reinterpret_cast<const float4*>(
                &stage[c & (NBUF - 1)][tid * 4]);
            const unsigned base = (unsigned)(v4 * 4);
            online_step(mA, sA, a0, a1, a2, ai0, ai1, ai2, x.x, base + 0u);
            online_step(mB, sB, b0, b1, b2, bi0, bi1, bi2, x.y, base + 1u);
            online_step(mA, sA, a0, a1, a2, ai0, ai1, ai2, x.z, base + 2u);
            online_step(mB, sB, b0, b1, b2, bi0, bi1, bi2, x.w, base + 3u);
        }
    };

    issue(0); issue(1); issue(2);

    for (int c = 0; c < NCHUNK - 2; ++c) {
        asm volatile("s_wait_asynccnt 0x2" ::: "memory"); // chunk c landed
        __syncthreads();            // buffer (c+3)&3 now safe to overwrite
        if (c + 3 < NCHUNK) issue(c + 3);
        process(c);
    }
    asm volatile("s_wait_asynccnt 0x1" ::: "memory");
    __syncthreads();
    process(NCHUNK - 2);
    asm volatile("s_wait_asynccnt 0x0" ::: "memory");
    __syncthreads();
    process(NCHUNK - 1);

    // Merge set B into set A (index-aware tie-break preserved).
    {
        const float mm = fmaxf(mA, mB);
        sA = sA * __expf(mA - mm) + sB * __expf(mB - mm);
        mA = mm;
        ins_top3(a0, a1, a2, ai0, ai1, ai2, b0, bi0);
        ins_top3(a0, a1, a2, ai0, ai1, ai2, b1, bi1);
        ins_top3(a0, a1, a2, ai0, ai1, ai2, b2, bi2);
    }

    // Cross-thread tree reduction in LDS.
    red_m[tid] = mA; red_s[tid] = sA;
    red_v[0][tid] = a0; red_v[1][tid] = a1; red_v[2][tid] = a2;
    red_i[0][tid] = ai0; red_i[1][tid] = ai1; red_i[2][tid] = ai2;
    for (int off = THREADS / 2; off > 0; off >>= 1) {
        __syncthreads();
        if (tid < off) {
            const int o = tid + off;
            const float ma = red_m[tid], mb = red_m[o];
            const float mm = fmaxf(ma, mb);
            red_s[tid] = red_s[tid] * __expf(ma - mm) +
                         red_s[o]   * __expf(mb - mm);
            red_m[tid] = mm;
            float t0 = red_v[0][tid], t1 = red_v[1][tid], t2 = red_v[2][tid];
            unsigned j0 = red_i[0][tid], j1 = red_i[1][tid], j2 = red_i[2][tid];
            ins_top3(t0, t1, t2, j0, j1, j2, red_v[0][o], red_i[0][o]);
            ins_top3(t0, t1, t2, j0, j1, j2, red_v[1][o], red_i[1][o]);
            ins_top3(t0, t1, t2, j0, j1, j2, red_v[2][o], red_i[2][o]);
            red_v[0][tid] = t0; red_v[1][tid] = t1; red_v[2][tid] = t2;
            red_i[0][tid] = j0; red_i[1][tid] = j1; red_i[2][tid] = j2;
        }
    }
    __syncthreads();

    if (tid == 0) {
        const float mm = red_m[0], ss = red_s[0];
        float* rec = ws + (size_t)row * 8;
        unsigned* ru = (unsigned*)rec;
#pragma unroll
        for (int j = 0; j < KK; ++j) {
            rec[j]    = logf(expf(red_v[j][0] - mm) / ss + EPSF);
            ru[3 + j] = red_i[j][0];
        }
    }
}

// Stage 2: one wave per batch. Serial K=3 beam scan over 128 per-step
// records (9 candidates/step; exact because every beam adds the same lp
// vector, so the global top-3 lies in {score[k] + lp_topj}), then
// parallel backtrack by lanes 0..2.
__global__ __launch_bounds__(32) void beam_kernel(
    const float* __restrict__ ws, float* __restrict__ out) {
    const int b    = blockIdx.x;
    const int lane = threadIdx.x;
    __shared__ int par[LL][KK];
    __shared__ int tok[LL][KK];
    __shared__ float fin[KK];

    if (lane == 0) {
        float scs[KK] = {0.0f, NEGF, NEGF};
        for (int l = 0; l < LL; ++l) {
            const float* rec = ws + (size_t)(b * LL + l) * 8;
            const unsigned* ru = (const unsigned*)rec;
            float lp[KK]; unsigned ti[KK];
            for (int j = 0; j < KK; ++j) { lp[j] = rec[j]; ti[j] = ru[3 + j]; }

            float cv[9]; unsigned cf[9]; int cp[9], ct[9];
            int n = 0;
            for (int k = 0; k < KK; ++k)
                for (int j = 0; j < KK; ++j) {
                    cv[n] = scs[k] + lp[j];
                    cf[n] = (unsigned)k * (unsigned)VV + ti[j];
                    cp[n] = k; ct[n] = (int)ti[j]; ++n;
                }
            unsigned usedMask = 0u;
            float ns[KK];
            for (int slot = 0; slot < KK; ++slot) {
                int best = -1;
                for (int t = 0; t < 9; ++t) {
                    if (usedMask & (1u << t)) continue;
                    if (best < 0 || cv[t] > cv[best] ||
                        (cv[t] == cv[best] && cf[t] < cf[best]))
                        best = t;
                }
                usedMask |= (1u << best);
                ns[slot] = cv[best];
                par[l][slot] = cp[best];
                tok[l][slot] = ct[best];
            }
            for (int k = 0; k < KK; ++k) scs[k] = ns[k];
        }
        for (int k = 0; k < KK; ++k) fin[k] = scs[k];
    }
    __syncthreads();

    if (lane < KK) {
        int beam = lane;
        for (int l = LL - 1; l >= 0; --l) {
            out[((size_t)b * LL + l) * KK + lane] = (float)tok[l][beam];
            beam = par[l][beam];
        }
        out[(size_t)BB * LL * KK + b * KK + lane] = fin[lane];
    }
}

extern "C" void kernel_launch(void* const* d_in, const int* in_sizes, int n_in,
                              void* d_out, int out_size, void* d_ws, size_t ws_size,
                              hipStream_t stream) {
    (void)in_sizes; (void)n_in; (void)out_size; (void)ws_size;
    const float* logits = (const float*)d_in[0];
    float* out = (float*)d_out;
    float* ws  = (float*)d_ws;   // 4096 rows x 8 dwords = 128 KB

    rowstats_kernel<<<BB * LL, THREADS, 0, stream>>>(logits, ws);
    beam_kernel<<<BB, 32, 0, stream>>>(ws, out);
}